// NNLS_47321949667728
// MI455X (gfx1250) — compile-verified
//
#include <hip/hip_runtime.h>

#define KDIM 1024
#define BDIM 8192
#define BM 128
#define BN 128
#define BK 32
#define LDSP 36   // padded LDS row stride (floats): keeps 16B alignment, breaks bank conflicts

typedef __attribute__((ext_vector_type(2))) float v2f;
typedef __attribute__((ext_vector_type(4))) float v4f;
typedef __attribute__((ext_vector_type(8))) float v8f;

// C[128x128] tile of th1 @ Y, fused NNLS-PGD epilogue.
// th1: [K,K] row-major, Y: [K,B] row-major.
// out layout (floats): [0, K*B)       = Y_new
//                      [K*B, 2K*B)    = new_X
//                      [2K*B]         = (float)(k+1)   (written by scalar kernel)
//                      [2K*B+1, 3K*B+1) = weight passthrough
__global__ __launch_bounds__(256) void nnls_pgd_gemm(
    const float* __restrict__ th1, const float* __restrict__ Y,
    const float* __restrict__ X_old, const int* __restrict__ kptr,
    const float* __restrict__ weight, float* __restrict__ out)
{
    __shared__ __align__(16) float As[BM][LDSP];   // A tile, row-major [m][k]
    __shared__ __align__(16) float Bt[BN][LDSP];   // B tile, transposed [n][k]

    const int tid  = threadIdx.x;
    const int lane = tid & 31;
    const int wave = tid >> 5;      // 0..7
    const int wm   = wave >> 1;     // 0..3  -> 32 rows each
    const int wn   = wave & 1;      // 0..1  -> 64 cols each
    const int half = lane >> 4;     // 0..1
    const int l16  = lane & 15;

    const int m0 = blockIdx.y * BM;
    const int n0 = blockIdx.x * BN;

    v8f acc[2][4];
    v8f zerov = {};
#pragma unroll
    for (int mt = 0; mt < 2; ++mt)
#pragma unroll
        for (int nt = 0; nt < 4; ++nt)
            acc[mt][nt] = zerov;

    for (int kk = 0; kk < KDIM; kk += BK) {
        // ---- stage A tile: 128 rows x 32 k, float4 per thread x4 ----
        {
            const int r = tid >> 3;            // 0..31
            const int c = (tid & 7) << 2;      // 0,4,...,28
#pragma unroll
            for (int rr = 0; rr < BM; rr += 32) {
                v4f v = *(const v4f*)&th1[(size_t)(m0 + r + rr) * KDIM + kk + c];
                *(v4f*)&As[r + rr][c] = v;     // 16B aligned: LDSP%4==0, c%4==0
            }
        }
        // ---- stage B tile transposed: 32 k-rows x 128 cols -> Bt[n][k] ----
        {
            const int r = tid >> 5;            // 0..7
            const int c = (tid & 31) << 2;     // 0..124
#pragma unroll
            for (int rr = 0; rr < BK; rr += 8) {
                v4f v = *(const v4f*)&Y[(size_t)(kk + r + rr) * BDIM + n0 + c];
                Bt[c + 0][r + rr] = v.x;
                Bt[c + 1][r + rr] = v.y;
                Bt[c + 2][r + rr] = v.z;
                Bt[c + 3][r + rr] = v.w;
            }
        }
        __syncthreads();

        // ---- compute: 8 k-steps of 4, 8 WMMAs each ----
#pragma unroll
        for (int ks = 0; ks < BK; ks += 4) {
            const int kb = ks + 2 * half;      // v[j] = elem at k = kb + j
            v2f a[2], b[4];
#pragma unroll
            for (int mt = 0; mt < 2; ++mt)
                a[mt] = *(const v2f*)&As[wm * 32 + mt * 16 + l16][kb];
#pragma unroll
            for (int nt = 0; nt < 4; ++nt)
                b[nt] = *(const v2f*)&Bt[wn * 64 + nt * 16 + l16][kb];
#pragma unroll
            for (int mt = 0; mt < 2; ++mt)
#pragma unroll
                for (int nt = 0; nt < 4; ++nt)
                    acc[mt][nt] = __builtin_amdgcn_wmma_f32_16x16x4_f32(
                        false, a[mt], false, b[nt],
                        (short)0, acc[mt][nt], false, false);
        }
        __syncthreads();
    }

    // ---- fused epilogue: relu + Nesterov momentum + 3 outputs ----
    const float kf  = (float)(*kptr);
    const float mom = (kf - 1.0f) / (kf + 2.0f);
    const size_t KB = (size_t)KDIM * BDIM;
    float* outY = out;
    float* outX = out + KB;
    float* outW = out + 2 * KB + 1;

#pragma unroll
    for (int mt = 0; mt < 2; ++mt) {
#pragma unroll
        for (int nt = 0; nt < 4; ++nt) {
            const int col = n0 + wn * 64 + nt * 16 + l16;
#pragma unroll
            for (int r = 0; r < 8; ++r) {
                const int row = m0 + wm * 32 + mt * 16 + half * 8 + r;
                const size_t idx = (size_t)row * BDIM + col;
                const float w  = weight[idx];
                const float xo = X_old[idx];
                float nx = acc[mt][nt][r] + w;
                nx = nx > 0.0f ? nx : 0.0f;
                outY[idx] = nx + mom * (nx - xo);
                outX[idx] = nx;
                outW[idx] = w;
            }
        }
    }
}

__global__ void nnls_pgd_scalar(const int* __restrict__ kptr, float* __restrict__ slot)
{
    if (threadIdx.x == 0 && blockIdx.x == 0)
        slot[0] = (float)(*kptr + 1);
}

extern "C" void kernel_launch(void* const* d_in, const int* in_sizes, int n_in,
                              void* d_out, int out_size, void* d_ws, size_t ws_size,
                              hipStream_t stream) {
    const float* th1   = (const float*)d_in[0];
    const float* Y     = (const float*)d_in[1];
    const float* X_old = (const float*)d_in[2];
    const int*   kptr  = (const int*)d_in[3];
    const float* weight= (const float*)d_in[4];
    float* out = (float*)d_out;

    dim3 grid(BDIM / BN, KDIM / BM);   // 64 x 8 = 512 workgroups
    nnls_pgd_gemm<<<grid, 256, 0, stream>>>(th1, Y, X_old, kptr, weight, out);

    nnls_pgd_scalar<<<1, 1, 0, stream>>>(kptr, out + (size_t)2 * KDIM * BDIM);
}